// DegreeOnlyFiltration_3624952397844
// MI455X (gfx1250) — compile-verified
//
#include <hip/hip_runtime.h>
#include <cstdint>

// Segmented max + broadcast divide.
//   out[k] = deg[k] / max(deg[seg(k)])
// Memory-bound: 64 MB read + 64 MB write -> ~5.5 us floor at 23.3 TB/s.
// Fast path streams each 8 KB segment into LDS with the CDNA5 async
// global->LDS mover, reads HBM exactly once, and writes NT stores.

typedef float v4f __attribute__((ext_vector_type(4)));

#define BLOCK    256
#define SEG_TILE 2048   // floats in the fast-path segment (8 KB)
#define NWAVES   (BLOCK / 32)

__global__ __launch_bounds__(BLOCK)
void seg_max_div_kernel(const float* __restrict__ deg,
                        const long long* __restrict__ pos,
                        float* __restrict__ out)
{
    __shared__ __align__(16) float tile[SEG_TILE];
    __shared__ float wred[NWAVES];

    const int g    = blockIdx.x;
    const int tid  = threadIdx.x;
    const int wid  = tid >> 5;
    const int lane = tid & 31;

    const long long start = pos[g];
    const long long end   = pos[g + 1];
    const long long len   = end - start;

    if (len == (long long)SEG_TILE && ((start & 3LL) == 0LL)) {
        // ---------- fast path: async-copy whole segment into LDS ----------
        const uint64_t base    = (uint64_t)(uintptr_t)deg;
        const uint32_t segByte = (uint32_t)(start * 4LL);
        const uint32_t ldsBase = (uint32_t)(uintptr_t)(&tile[0]);

        {
            uint32_t voff0 = segByte + (uint32_t)tid * 16u;
            uint32_t ldso0 = ldsBase + (uint32_t)tid * 16u;
            uint32_t voff1 = voff0 + 4096u;
            uint32_t ldso1 = ldso0 + 4096u;
            asm volatile("global_load_async_to_lds_b128 %0, %1, %2"
                         :: "v"(ldso0), "v"(voff0), "s"(base) : "memory");
            asm volatile("global_load_async_to_lds_b128 %0, %1, %2"
                         :: "v"(ldso1), "v"(voff1), "s"(base) : "memory");
        }
        // wait for this wave's async transfers, then make LDS visible WG-wide
        asm volatile("s_wait_asynccnt 0x0" ::: "memory");
        __syncthreads();

        // each thread owns float4 tiles j = tid and j = tid + BLOCK
        v4f a0 = *reinterpret_cast<const v4f*>(&tile[(size_t)tid * 4]);
        v4f a1 = *reinterpret_cast<const v4f*>(&tile[(size_t)(tid + BLOCK) * 4]);

        float m = fmaxf(fmaxf(fmaxf(a0.x, a0.y), fmaxf(a0.z, a0.w)),
                        fmaxf(fmaxf(a1.x, a1.y), fmaxf(a1.z, a1.w)));

        // wave32 butterfly max
        #pragma unroll
        for (int off = 16; off >= 1; off >>= 1)
            m = fmaxf(m, __shfl_xor(m, off, 32));
        if (lane == 0) wred[wid] = m;
        __syncthreads();

        float bm = wred[0];
        #pragma unroll
        for (int w = 1; w < NWAVES; ++w) bm = fmaxf(bm, wred[w]);

        const float inv = 1.0f / bm;   // one IEEE divide per thread
        v4f r0 = a0 * inv;
        v4f r1 = a1 * inv;

        v4f* o = reinterpret_cast<v4f*>(out + start);
        __builtin_nontemporal_store(r0, o + tid);
        __builtin_nontemporal_store(r1, o + tid + BLOCK);
    } else {
        // ---------- general path: arbitrary segment boundaries ----------
        float m = -__builtin_inff();
        for (long long i = start + tid; i < end; i += BLOCK)
            m = fmaxf(m, deg[i]);

        #pragma unroll
        for (int off = 16; off >= 1; off >>= 1)
            m = fmaxf(m, __shfl_xor(m, off, 32));
        if (lane == 0) wred[wid] = m;
        __syncthreads();

        float bm = wred[0];
        #pragma unroll
        for (int w = 1; w < NWAVES; ++w) bm = fmaxf(bm, wred[w]);

        for (long long i = start + tid; i < end; i += BLOCK)
            out[i] = deg[i] / bm;      // exact divide in the slow path
    }
}

extern "C" void kernel_launch(void* const* d_in, const int* in_sizes, int n_in,
                              void* d_out, int out_size, void* d_ws, size_t ws_size,
                              hipStream_t stream)
{
    const float*     deg = (const float*)d_in[0];
    const long long* pos = (const long long*)d_in[1];  // int64 boundaries
    float*           out = (float*)d_out;

    const int n_graphs = in_sizes[1] - 1;              // 8192 segments
    seg_max_div_kernel<<<n_graphs, BLOCK, 0, stream>>>(deg, pos, out);
}